// TokenEmbedding_59536836657397
// MI455X (gfx1250) — compile-verified
//
#include <hip/hip_runtime.h>

// Token embedding gather + LayerNorm, MI455X (gfx1250, wave32).
// tokens:[32768] i32, table:[50277,1024] f32, gamma/beta:[1024] f32
// out:[32768,1024] f32.  Pure HBM-bandwidth kernel (~268 MB => ~11.5us floor).
//
// Hot path: wave-per-row, 8x global_load_b128 clause (scalar saddr from the
// token id), in-wave butterfly reduction, non-temporal b128 stores.
// gamma/beta are staged into LDS once per block via the Tensor Data Mover
// (TENSORcnt-tracked global->LDS DMA, zero VGPR traffic).

typedef float        v4f   __attribute__((ext_vector_type(4)));
typedef unsigned int u32x4 __attribute__((ext_vector_type(4)));
typedef int          i32x4 __attribute__((ext_vector_type(4)));
typedef int          i32x8 __attribute__((ext_vector_type(8)));

#define N_EMBD 1024
#define VEC_PER_ROW (N_EMBD / 4)        // 256 float4 per row
#define VEC_PER_LANE (VEC_PER_ROW / 32) // 8 float4 per lane (wave32)
#define WAVES_PER_BLOCK 8
#define BLOCK_THREADS (WAVES_PER_BLOCK * 32)
#define LN_EPS 1e-5f

#define HAS_TDM (__has_builtin(__builtin_amdgcn_tensor_load_to_lds) && \
                 __has_builtin(__builtin_amdgcn_s_wait_tensorcnt))

#if HAS_TDM
// Build D# group 0 (ISA cdna5 §8.3): count=1, lds_addr[63:32],
// global_addr[120:64], type=2 ("image") in [127:126].
__device__ __forceinline__ u32x4 tdm_group0(unsigned lds_addr, const void* g) {
  const unsigned long long ga = (unsigned long long)(uintptr_t)g;
  u32x4 g0;
  g0.x = 1u;                                               // count=1, no gather
  g0.y = lds_addr;                                         // LDS byte address
  g0.z = (unsigned)ga;                                     // global_addr[31:0]
  g0.w = ((unsigned)(ga >> 32) & 0x01FFFFFFu) | (2u << 30);// addr[56:32]|type=2
  return g0;
}
#endif

__global__ __launch_bounds__(BLOCK_THREADS)
void emb_layernorm_kernel(const int* __restrict__ tokens,
                          const float* __restrict__ table,
                          const float* __restrict__ gamma,
                          const float* __restrict__ beta,
                          float* __restrict__ out,
                          int n_rows) {
  __shared__ float s_g[N_EMBD];
  __shared__ float s_b[N_EMBD];

#if HAS_TDM
  if (threadIdx.x < 32) {  // wave 0 issues the DMA (TDM ignores EXEC; one
                           // issue per wave, so gate to a single wave)
    // D# group 1 (ISA §8.4): 2D tensor, one line of 1024 fp32.
    i32x8 g1 = {};
    g1[0] = 0x00020000;            // data_size=2 -> 4-byte elements
    g1[1] = (int)(N_EMBD << 16);   // tensor_dim0 = 1024 (bits 79:48, low half)
    g1[2] = 0x00010000;            // tensor_dim1 = 1    (bits 111:80)
    g1[3] = (int)(N_EMBD << 16);   // tile_dim0   = 1024 (bits 127:112)
    g1[4] = 1;                     // tile_dim1=1, tile_dim2=0
    g1[5] = N_EMBD;                // tensor_dim0_stride = 1024
    const i32x4 z4 = {};           // groups 2/3 unused for 2D tensors
    const i32x8 z8 = {};           // extra group (clang-23 6-arg form), unused
    __builtin_amdgcn_tensor_load_to_lds(
        tdm_group0((unsigned)(uintptr_t)s_g, gamma), g1, z4, z4, z8, 0);
    __builtin_amdgcn_tensor_load_to_lds(
        tdm_group0((unsigned)(uintptr_t)s_b, beta), g1, z4, z4, z8, 0);
    __builtin_amdgcn_s_wait_tensorcnt(0);
  }
  __syncthreads();
#else
  {
    const int t = threadIdx.x;     // fallback: 256 threads x 1 float4 each
    ((v4f*)s_g)[t] = ((const v4f*)gamma)[t];
    ((v4f*)s_b)[t] = ((const v4f*)beta)[t];
  }
  __syncthreads();
#endif

  const int wave = threadIdx.x >> 5;
  const int lane = threadIdx.x & 31;
  const int row = blockIdx.x * WAVES_PER_BLOCK + wave;
  if (row >= n_rows) return;

  // Wave-uniform row index -> SGPR, so the token fetch and the gather base
  // address computation stay entirely on the scalar path.
  const int urow = __builtin_amdgcn_readfirstlane(row);
  const int tok  = tokens[urow];
  const v4f* __restrict__ src = (const v4f*)(table + (size_t)tok * N_EMBD);

  // 8 x global_load_b128 in one clause: each is a coalesced 512B wave
  // request; 8 loads in flight per lane + 32K resident waves hide HBM latency.
  v4f v[VEC_PER_LANE];
#pragma unroll
  for (int i = 0; i < VEC_PER_LANE; ++i)
    v[i] = src[lane + (i << 5)];

  // Per-lane partials over 32 elements.
  float s = 0.0f, ss = 0.0f;
#pragma unroll
  for (int i = 0; i < VEC_PER_LANE; ++i) {
    s += (v[i].x + v[i].y) + (v[i].z + v[i].w);
    ss = fmaf(v[i].x, v[i].x, ss);
    ss = fmaf(v[i].y, v[i].y, ss);
    ss = fmaf(v[i].z, v[i].z, ss);
    ss = fmaf(v[i].w, v[i].w, ss);
  }

  // wave32 butterfly reduction; the whole row lives in one wave.
#pragma unroll
  for (int off = 16; off >= 1; off >>= 1) {
    s  += __shfl_xor(s, off, 32);
    ss += __shfl_xor(ss, off, 32);
  }

  const float inv_n = 1.0f / (float)N_EMBD;
  const float mean = s * inv_n;
  const float var  = fmaf(ss, inv_n, -mean * mean); // E[x^2] - mean^2
  const float rstd = rsqrtf(var + LN_EPS);          // v_rsq_f32

  v4f* __restrict__ dst = (v4f*)(out + (size_t)row * N_EMBD);
  const v4f* __restrict__ g4 = (const v4f*)s_g;
  const v4f* __restrict__ b4 = (const v4f*)s_b;

#pragma unroll
  for (int i = 0; i < VEC_PER_LANE; ++i) {
    const int j = lane + (i << 5);
    const v4f g = g4[j];
    const v4f b = b4[j];
    v4f o;
    o.x = fmaf((v[i].x - mean) * rstd, g.x, b.x);
    o.y = fmaf((v[i].y - mean) * rstd, g.y, b.y);
    o.z = fmaf((v[i].z - mean) * rstd, g.z, b.z);
    o.w = fmaf((v[i].w - mean) * rstd, g.w, b.w);
    // Write-once streaming output: non-temporal b128 stores keep the 134 MB
    // of writes from evicting the 206 MB table out of the 192 MB L2, so
    // repeated-token gathers (~27% of rows) keep hitting L2.
    __builtin_nontemporal_store(o, &dst[j]);
  }
}

extern "C" void kernel_launch(void* const* d_in, const int* in_sizes, int n_in,
                              void* d_out, int out_size, void* d_ws, size_t ws_size,
                              hipStream_t stream) {
  const int*   tokens = (const int*)d_in[0];
  const float* table  = (const float*)d_in[1];
  const float* gamma  = (const float*)d_in[2];
  const float* beta   = (const float*)d_in[3];
  float*       out    = (float*)d_out;

  const int n_rows = in_sizes[0]; // 8 * 4096 = 32768 tokens
  const int blocks = (n_rows + WAVES_PER_BLOCK - 1) / WAVES_PER_BLOCK;

  emb_layernorm_kernel<<<blocks, BLOCK_THREADS, 0, stream>>>(
      tokens, table, gamma, beta, out, n_rows);
}